// adder2d_87608742904026
// MI455X (gfx1250) — compile-verified
//
#include <hip/hip_runtime.h>
#include <math.h>

// AdderNet conv on MI455X (gfx1250).
//
// out[n,f,oh,ow] = -sum_{c,kh,kw} |x[n,c,oh-1+kh,ow-1+kw] - w[f,c,kh,kw]|
//
// WMMA is mathematically inapplicable (|a-b| is not bilinear), so the optimal
// implementation hits the VALU floor of 2 ops/term (v_sub_f32 + v_add_f32
// with |src| modifier), keeps weights in SGPRs via uniform (scalar) loads,
// and streams a pre-padded x image (L2-resident) with branch-free b96 row
// loads + global_prefetch_b8 for the next channel.
//
// Round-2 refinement: 2 vertically-adjacent output pixels per thread.
// The two 3x3 windows share 2 of 3 rows -> per channel a thread loads 4 rows
// for 2 pixels (2 vmem/pixel instead of 3), VALU per body doubles to 288
// (~96% of issue slots are the SAD math), and the SGPR-resident weights are
// reused for both pixels (scalar-cache traffic per output halved).

#define N_  16
#define C_  64
#define H_  56
#define W_  56
#define F_  128
#define KK  9      // 3x3
#define HP  58     // padded
#define WP  58
#define FT  8      // filters per block (accumulator pairs per thread)
#define TROWS 4                       // thread-rows per block
#define PIXV  2                       // vertical pixels per thread
#define THREADS (TROWS * W_)          // 224 = 7 wave32
#define OROWS (TROWS * PIXV)          // 8 output rows per block

// ---------------------------------------------------------------------------
// Kernel 1: zero-padded copy of x into workspace: xp[n][c][58][58]
// ---------------------------------------------------------------------------
__global__ __launch_bounds__(256) void adder2d_pad_kernel(
    const float* __restrict__ x, float* __restrict__ xp, int total) {
  int idx = blockIdx.x * blockDim.x + threadIdx.x;
  if (idx >= total) return;
  int iwp = idx % WP;
  int t   = idx / WP;
  int ihp = t % HP;
  t /= HP;
  int c = t % C_;
  int n = t / C_;
  int ih = ihp - 1, iw = iwp - 1;
  float v = 0.0f;
  if ((unsigned)ih < (unsigned)H_ && (unsigned)iw < (unsigned)W_)
    v = x[((n * C_ + c) * H_ + ih) * W_ + iw];
  xp[idx] = v;
}

// ---------------------------------------------------------------------------
// Kernel 2 (main path): pre-padded input, 2 output pixels per thread.
// ---------------------------------------------------------------------------
__global__ __launch_bounds__(THREADS) void adder2d_main_kernel(
    const float* __restrict__ xp,      // [N][C][58][58] padded
    const float* __restrict__ w,       // [F][C][3][3]
    float* __restrict__ out) {         // [N][F][56][56]
  const int tid = threadIdx.x;
  const int ow  = tid % W_;
  const int r   = tid / W_;                        // 0..3
  const int oh0 = blockIdx.x * OROWS + r * PIXV;   // first of 2 pixels
  const int f0  = blockIdx.y * FT;                 // filter chunk base (uniform)
  const int n   = blockIdx.z;

  float acc0[FT], acc1[FT];
#pragma unroll
  for (int i = 0; i < FT; ++i) { acc0[i] = 0.0f; acc1[i] = 0.0f; }

  // Wave-uniform weight base: all weight reads below become s_load -> SGPRs.
  const float* wbase = w + (size_t)f0 * (C_ * KK);

#pragma unroll 1
  for (int c = 0; c < C_; ++c) {
    // 4 padded rows (oh0 .. oh0+3), 3 contiguous floats each -> 4x b96.
    const float* xr = xp + (((size_t)(n * C_ + c) * HP + oh0) * WP + ow);
    float a0 = xr[0],      a1 = xr[1],          a2 = xr[2];           // row 0
    float b0 = xr[WP],     b1 = xr[WP + 1],     b2 = xr[WP + 2];      // row 1
    float c0 = xr[2 * WP], c1 = xr[2 * WP + 1], c2 = xr[2 * WP + 2];  // row 2
    float d0 = xr[3 * WP], d1 = xr[3 * WP + 1], d2 = xr[3 * WP + 2];  // row 3

    // gfx1250 prefetch (global_prefetch_b8): pull next channel's rows into
    // cache while the 288-op VALU burst below executes.
    if (c + 1 < C_) {
      const float* xn = xr + (size_t)HP * WP;
      __builtin_prefetch(xn, 0, 1);
      __builtin_prefetch(xn + WP, 0, 1);
      __builtin_prefetch(xn + 2 * WP, 0, 1);
      __builtin_prefetch(xn + 3 * WP, 0, 1);
    }

    const float* wc = wbase + c * KK;   // uniform
#pragma unroll
    for (int i = 0; i < FT; ++i) {
      const float* wi = wc + i * (C_ * KK);   // uniform -> SGPRs
      float w0 = wi[0], w1 = wi[1], w2 = wi[2];
      float w3 = wi[3], w4 = wi[4], w5 = wi[5];
      float w6 = wi[6], w7 = wi[7], w8 = wi[8];
      // pixel oh0: rows a,b,c
      acc0[i] += fabsf(a0 - w0) + fabsf(a1 - w1) + fabsf(a2 - w2) +
                 fabsf(b0 - w3) + fabsf(b1 - w4) + fabsf(b2 - w5) +
                 fabsf(c0 - w6) + fabsf(c1 - w7) + fabsf(c2 - w8);
      // pixel oh0+1: rows b,c,d (same SGPR weights, shared x rows)
      acc1[i] += fabsf(b0 - w0) + fabsf(b1 - w1) + fabsf(b2 - w2) +
                 fabsf(c0 - w3) + fabsf(c1 - w4) + fabsf(c2 - w5) +
                 fabsf(d0 - w6) + fabsf(d1 - w7) + fabsf(d2 - w8);
    }
  }

#pragma unroll
  for (int i = 0; i < FT; ++i) {
    size_t o = (((size_t)n * F_ + (f0 + i)) * H_ + oh0) * W_ + ow;
    out[o]      = -acc0[i];
    out[o + W_] = -acc1[i];
  }
}

// ---------------------------------------------------------------------------
// Fallback (ws too small): bounds-checked reads directly from x, 1 px/thread.
// ---------------------------------------------------------------------------
__global__ __launch_bounds__(THREADS) void adder2d_fallback_kernel(
    const float* __restrict__ x, const float* __restrict__ w,
    float* __restrict__ out) {
  const int tid = threadIdx.x;
  const int ow  = tid % W_;
  const int r   = tid / W_;
  const int oh  = blockIdx.x * TROWS + r;
  const int f0  = blockIdx.y * FT;
  const int n   = blockIdx.z;

  float acc[FT];
#pragma unroll
  for (int i = 0; i < FT; ++i) acc[i] = 0.0f;

  const float* wbase = w + (size_t)f0 * (C_ * KK);

#pragma unroll 1
  for (int c = 0; c < C_; ++c) {
    const int ih0 = oh - 1, iw0 = ow - 1;
    const float* xc = x + ((size_t)(n * C_ + c) * H_) * W_;
    auto ld = [&](int ih, int iw) -> float {
      return ((unsigned)ih < (unsigned)H_ && (unsigned)iw < (unsigned)W_)
                 ? xc[ih * W_ + iw] : 0.0f;
    };
    float x00 = ld(ih0, iw0),     x01 = ld(ih0, iw0 + 1),     x02 = ld(ih0, iw0 + 2);
    float x10 = ld(ih0 + 1, iw0), x11 = ld(ih0 + 1, iw0 + 1), x12 = ld(ih0 + 1, iw0 + 2);
    float x20 = ld(ih0 + 2, iw0), x21 = ld(ih0 + 2, iw0 + 1), x22 = ld(ih0 + 2, iw0 + 2);

    const float* wc = wbase + c * KK;
#pragma unroll
    for (int i = 0; i < FT; ++i) {
      const float* wi = wc + i * (C_ * KK);
      acc[i] += fabsf(x00 - wi[0]) + fabsf(x01 - wi[1]) + fabsf(x02 - wi[2]) +
                fabsf(x10 - wi[3]) + fabsf(x11 - wi[4]) + fabsf(x12 - wi[5]) +
                fabsf(x20 - wi[6]) + fabsf(x21 - wi[7]) + fabsf(x22 - wi[8]);
    }
  }

#pragma unroll
  for (int i = 0; i < FT; ++i)
    out[(((size_t)n * F_ + (f0 + i)) * H_ + oh) * W_ + ow] = -acc[i];
}

// ---------------------------------------------------------------------------
extern "C" void kernel_launch(void* const* d_in, const int* in_sizes, int n_in,
                              void* d_out, int out_size, void* d_ws, size_t ws_size,
                              hipStream_t stream) {
  const float* x = (const float*)d_in[0];   // [16][64][56][56] f32
  const float* w = (const float*)d_in[1];   // [128][64][3][3]  f32
  float* out = (float*)d_out;               // [16][128][56][56] f32

  const size_t pad_elems = (size_t)N_ * C_ * HP * WP;       // 3,444,736
  const size_t pad_bytes = pad_elems * sizeof(float);       // ~13.8 MB

  if (ws_size >= pad_bytes) {
    float* xp = (float*)d_ws;
    int total = (int)pad_elems;
    adder2d_pad_kernel<<<(total + 255) / 256, 256, 0, stream>>>(x, xp, total);
    dim3 grid(H_ / OROWS, F_ / FT, N_);                     // 7 x 16 x 16
    adder2d_main_kernel<<<grid, THREADS, 0, stream>>>(xp, w, out);
  } else {
    dim3 grid(H_ / TROWS, F_ / FT, N_);                     // 14 x 16 x 16
    adder2d_fallback_kernel<<<grid, THREADS, 0, stream>>>(x, w, out);
  }
}